// Cross_Attention_Global_5952824672973
// MI455X (gfx1250) — compile-verified
//
#include <hip/hip_runtime.h>
#include <hip/hip_bf16.h>

typedef __attribute__((ext_vector_type(2))) float v2f;
typedef __attribute__((ext_vector_type(8))) float v8f;
typedef int v4i __attribute__((vector_size(4 * sizeof(int))));

#define B_DIM 4
#define N_DIM 16384
#define C_DIM 64
#define H_DIM 8
#define EPS_IN 1e-5f

#ifndef __has_builtin
#define __has_builtin(x) 0
#endif

#if __has_builtin(__builtin_amdgcn_global_load_async_to_lds_b128)
#define HAS_ASYNC_LDS 1
#else
#define HAS_ASYNC_LDS 0
#endif

#if __has_builtin(__builtin_amdgcn_s_wait_asynccnt)
#define WAIT_ASYNC(n) __builtin_amdgcn_s_wait_asynccnt(n)
#elif HAS_ASYNC_LDS
#define WAIT_ASYNC(n) asm volatile("s_wait_asynccnt %0" ::"i"(n) : "memory")
#else
#define WAIT_ASYNC(n) ((void)0)
#endif

// addrspace conversions via ptrtoint/inttoptr; pointee type int32x4 per the
// builtin's probed signature (AS1 v4i* source, AS3 v4i* LDS destination).
#define GLB_AS1(p) \
  ((__attribute__((address_space(1))) v4i*)(unsigned long long)(p))
#define LDS_AS3(p) \
  ((__attribute__((address_space(3))) v4i*)(unsigned int)(unsigned long long)(p))

// Copy 4096 floats (16 KB) global -> LDS; 512 threads x 2 x 16B each.
static __device__ __forceinline__ void stage4k(const float* __restrict__ g,
                                               float* l, int t) {
#if HAS_ASYNC_LDS
#pragma unroll
  for (int i = 0; i < 2; ++i) {
    __builtin_amdgcn_global_load_async_to_lds_b128(
        GLB_AS1(g + (size_t)(t + i * 512) * 4),
        LDS_AS3(l + (size_t)(t + i * 512) * 4), 0, 0);
  }
#else
  const float4* s4 = (const float4*)g;
  float4* d4 = (float4*)l;
  d4[t] = s4[t];
  d4[t + 512] = s4[t + 512];
#endif
}

// D = A(16x4 f32) * B(4x16 f32) + C, full-fp32 WMMA (CDNA5 V_WMMA_F32_16X16X4_F32)
static __device__ __forceinline__ v8f wmma4(v2f a, v2f b, v8f c) {
  return __builtin_amdgcn_wmma_f32_16x16x4_f32(false, a, false, b, (short)0, c,
                                               false, false);
}

__global__ void k_zero(float* __restrict__ g, int n) {
  int i = blockIdx.x * blockDim.x + threadIdx.x;
  if (i < n) g[i] = 0.0f;
}

// G_b = emb_b^T @ emb_b (per-batch 64x64 Gram), streamed over N with
// double-buffered async global->LDS staging.
__global__ void __launch_bounds__(512) k_gram(const float* __restrict__ emb,
                                              float* __restrict__ G) {
  __shared__ float sE[2][4096];  // two 64-row chunks
  const int b   = blockIdx.x >> 6;  // 64 blocks per batch, 256 rows each
  const int blk = blockIdx.x & 63;
  const int t = threadIdx.x;
  const int lane = t & 31, wave = t >> 5;
  const int m = lane & 15;
  const int khalf = (lane < 16) ? 0 : 2;
  const int rowoff = (lane < 16) ? 0 : 8;
  const int ci = wave >> 2, di = wave & 3;  // 16x16 tile of G
  const float* base = emb + ((size_t)b * N_DIM + blk * 256) * C_DIM;

  stage4k(base, sE[0], t);  // chunk 0 in flight
  v8f acc = {};

#define GRAM_STEP(c, W, NEXT)                                         \
  {                                                                   \
    if (NEXT) stage4k(base + ((c) + 1) * 4096, sE[((c) + 1) & 1], t); \
    WAIT_ASYNC(W);                                                    \
    __syncthreads();                                                  \
    const float* sb = sE[(c) & 1];                                    \
    _Pragma("unroll") for (int kk = 0; kk < 16; ++kk) {               \
      const int kb = kk * 4 + khalf;                                  \
      v2f a, bv;                                                      \
      a.x  = sb[kb * 64 + ci * 16 + m];                               \
      a.y  = sb[(kb + 1) * 64 + ci * 16 + m];                         \
      bv.x = sb[kb * 64 + di * 16 + m];                               \
      bv.y = sb[(kb + 1) * 64 + di * 16 + m];                         \
      acc = wmma4(a, bv, acc);                                        \
    }                                                                 \
    __syncthreads();                                                  \
  }

  GRAM_STEP(0, 2, 1)
  GRAM_STEP(1, 2, 1)
  GRAM_STEP(2, 2, 1)
  GRAM_STEP(3, 0, 0)
#undef GRAM_STEP

  float* Gb = G + b * 4096;
#pragma unroll
  for (int v = 0; v < 8; ++v) {
    const int row = ci * 16 + v + rowoff;
    const int col = di * 16 + m;
    atomicAdd(&Gb[row * 64 + col], acc[v]);
  }
}

// Per (b,h): attn = IN+softmax(Wq_h @ G_b @ Wk_h^T); M2 = attn @ Wv_h,
// stored interleaved as M2r[b][(c*H+h)*64 + e] for the downstream GEMM.
__global__ void __launch_bounds__(512) k_attn(const float* __restrict__ G,
                                              const float* __restrict__ Wq,
                                              const float* __restrict__ Wk,
                                              const float* __restrict__ Wv,
                                              float* __restrict__ M2) {
  __shared__ float sG[4096], sT[4096], sA[4096];
  __shared__ float rs[512], rq[512];
  const int b = blockIdx.x >> 3;
  const int h = blockIdx.x & 7;
  const int t = threadIdx.x;
  const int lane = t & 31, wave = t >> 5;
  const int m = lane & 15;
  const int khalf = (lane < 16) ? 0 : 2;
  const int rowoff = (lane < 16) ? 0 : 8;
  const int ti = wave >> 2, tj = wave & 3;

  {
    const float4* src = (const float4*)(G + b * 4096);
    float4* dst = (float4*)sG;
    dst[t] = src[t];
    dst[t + 512] = src[t + 512];
  }
  __syncthreads();

  const float* Wqh = Wq + h * 64 * 64;
  const float* Wkh = Wk + h * 64 * 64;
  const float* Wvh = Wv + h * 64 * 64;

  // T1 = G @ Wk_h^T
  {
    v8f acc = {};
#pragma unroll
    for (int kk = 0; kk < 16; ++kk) {
      const int kb = kk * 4 + khalf;
      v2f a, bv;
      a.x  = sG[(ti * 16 + m) * 64 + kb];
      a.y  = sG[(ti * 16 + m) * 64 + kb + 1];
      bv.x = Wkh[(tj * 16 + m) * 64 + kb];
      bv.y = Wkh[(tj * 16 + m) * 64 + kb + 1];
      acc = wmma4(a, bv, acc);
    }
#pragma unroll
    for (int v = 0; v < 8; ++v)
      sT[(ti * 16 + v + rowoff) * 64 + tj * 16 + m] = acc[v];
  }
  __syncthreads();

  // attn = Wq_h @ T1
  {
    v8f acc = {};
#pragma unroll
    for (int kk = 0; kk < 16; ++kk) {
      const int kb = kk * 4 + khalf;
      v2f a, bv;
      a.x  = Wqh[(ti * 16 + m) * 64 + kb];
      a.y  = Wqh[(ti * 16 + m) * 64 + kb + 1];
      bv.x = sT[kb * 64 + tj * 16 + m];
      bv.y = sT[(kb + 1) * 64 + tj * 16 + m];
      acc = wmma4(a, bv, acc);
    }
#pragma unroll
    for (int v = 0; v < 8; ++v)
      sA[(ti * 16 + v + rowoff) * 64 + tj * 16 + m] = acc[v];
  }
  __syncthreads();

  // InstanceNorm2d (biased variance) over the whole 64x64 map
  float s = 0.f, q = 0.f;
#pragma unroll
  for (int i = 0; i < 8; ++i) {
    const float x = sA[t * 8 + i];
    s += x;
    q += x * x;
  }
  rs[t] = s;
  rq[t] = q;
  __syncthreads();
  for (int off = 256; off > 0; off >>= 1) {
    if (t < off) {
      rs[t] += rs[t + off];
      rq[t] += rq[t + off];
    }
    __syncthreads();
  }
  const float mean = rs[0] * (1.0f / 4096.0f);
  const float var  = rq[0] * (1.0f / 4096.0f) - mean * mean;
  const float inv  = rsqrtf(var + EPS_IN);
#pragma unroll
  for (int i = 0; i < 8; ++i)
    sA[t * 8 + i] = (sA[t * 8 + i] - mean) * inv;
  __syncthreads();

  // softmax over last dim (each of 64 rows)
  if (t < 64) {
    float mx = -1e30f;
    for (int j = 0; j < 64; ++j) mx = fmaxf(mx, sA[t * 64 + j]);
    float se = 0.f;
    for (int j = 0; j < 64; ++j) se += __expf(sA[t * 64 + j] - mx);
    const float r = 1.0f / se;
    for (int j = 0; j < 64; ++j) sA[t * 64 + j] = __expf(sA[t * 64 + j] - mx) * r;
  }
  __syncthreads();

  // M2 = attn @ Wv_h ; store interleaved
  {
    v8f acc = {};
#pragma unroll
    for (int kk = 0; kk < 16; ++kk) {
      const int kb = kk * 4 + khalf;
      v2f a, bv;
      a.x  = sA[(ti * 16 + m) * 64 + kb];
      a.y  = sA[(ti * 16 + m) * 64 + kb + 1];
      bv.x = Wvh[kb * 64 + tj * 16 + m];
      bv.y = Wvh[(kb + 1) * 64 + tj * 16 + m];
      acc = wmma4(a, bv, acc);
    }
    float* M2b = M2 + (size_t)b * 512 * 64;
#pragma unroll
    for (int v = 0; v < 8; ++v) {
      const int c = ti * 16 + v + rowoff;
      const int e = tj * 16 + m;
      M2b[(c * H_DIM + h) * 64 + e] = acc[v];
    }
  }
}

// P_b = Wo[64x512] @ M2r_b[512x64]
__global__ void __launch_bounds__(512) k_pmat(const float* __restrict__ M2,
                                              const float* __restrict__ Wo,
                                              float* __restrict__ P) {
  const int b = blockIdx.x;
  const int t = threadIdx.x;
  const int lane = t & 31, wave = t >> 5;
  const int m = lane & 15;
  const int khalf = (lane < 16) ? 0 : 2;
  const int rowoff = (lane < 16) ? 0 : 8;
  const int ti = wave >> 2, tj = wave & 3;
  const float* M2b = M2 + (size_t)b * 512 * 64;

  v8f acc = {};
  for (int kk = 0; kk < 128; ++kk) {
    const int kb = kk * 4 + khalf;
    v2f a, bv;
    a.x  = Wo[(ti * 16 + m) * 512 + kb];
    a.y  = Wo[(ti * 16 + m) * 512 + kb + 1];
    bv.x = M2b[kb * 64 + tj * 16 + m];
    bv.y = M2b[(kb + 1) * 64 + tj * 16 + m];
    acc = wmma4(a, bv, acc);
  }
  float* Pb = P + b * 4096;
#pragma unroll
  for (int v = 0; v < 8; ++v)
    Pb[(ti * 16 + v + rowoff) * 64 + tj * 16 + m] = acc[v];
}

// out_b = emb_b @ P_b^T, streamed over N: async-staged P_b + double-buffered
// emb chunks; non-temporal output stores.
__global__ void __launch_bounds__(512) k_out(const float* __restrict__ emb,
                                             const float* __restrict__ P,
                                             float* __restrict__ out) {
  __shared__ float sP[4096];
  __shared__ float sE[2][4096];
  const int b   = blockIdx.x >> 6;  // 64 blocks per batch, 256 rows each
  const int blk = blockIdx.x & 63;
  const int n0  = blk * 256;
  const int t = threadIdx.x;
  const int lane = t & 31, wave = t >> 5;
  const int m = lane & 15;
  const int khalf = (lane < 16) ? 0 : 2;
  const int rowoff = (lane < 16) ? 0 : 8;
  const int nt = wave >> 2, ct = wave & 3;
  const float* ebase = emb + ((size_t)b * N_DIM + n0) * C_DIM;

  stage4k(P + b * 4096, sP, t);  // issued first: in-order ahead of chunks
  stage4k(ebase, sE[0], t);      // chunk 0
  WAIT_ASYNC(2);                 // sP landed (in-order); chunk0 may be in flight
  __syncthreads();

  // preload B = P^T fragment for this wave's output-column tile (32 VGPRs)
  v2f brg[16];
#pragma unroll
  for (int kk = 0; kk < 16; ++kk) {
    const int kb = kk * 4 + khalf;
    brg[kk].x = sP[(ct * 16 + m) * 64 + kb];
    brg[kk].y = sP[(ct * 16 + m) * 64 + kb + 1];
  }

  float* ob = out + ((size_t)b * N_DIM + n0) * C_DIM;

#define OUT_STEP(c, W, NEXT)                                            \
  {                                                                     \
    if (NEXT) stage4k(ebase + ((c) + 1) * 4096, sE[((c) + 1) & 1], t);  \
    WAIT_ASYNC(W);                                                      \
    __syncthreads();                                                    \
    const float* sb = sE[(c) & 1];                                      \
    v8f acc = {};                                                       \
    _Pragma("unroll") for (int kk = 0; kk < 16; ++kk) {                 \
      const int kb = kk * 4 + khalf;                                    \
      v2f a;                                                            \
      a.x = sb[(nt * 16 + m) * 64 + kb];                                \
      a.y = sb[(nt * 16 + m) * 64 + kb + 1];                            \
      acc = wmma4(a, brg[kk], acc);                                     \
    }                                                                   \
    _Pragma("unroll") for (int v = 0; v < 8; ++v)                       \
        __builtin_nontemporal_store(                                    \
            acc[v],                                                     \
            &ob[((c) * 64 + nt * 16 + v + rowoff) * 64 + ct * 16 + m]); \
    __syncthreads();                                                    \
  }

  OUT_STEP(0, 2, 1)
  OUT_STEP(1, 2, 1)
  OUT_STEP(2, 2, 1)
  OUT_STEP(3, 0, 0)
#undef OUT_STEP
}

extern "C" void kernel_launch(void* const* d_in, const int* in_sizes, int n_in,
                              void* d_out, int out_size, void* d_ws,
                              size_t ws_size, hipStream_t stream) {
  (void)in_sizes; (void)n_in; (void)out_size; (void)ws_size;
  const float* emb = (const float*)d_in[0];
  const float* Wq  = (const float*)d_in[1];
  const float* Wk  = (const float*)d_in[2];
  const float* Wv  = (const float*)d_in[3];
  const float* Wo  = (const float*)d_in[4];
  float* out = (float*)d_out;

  float* ws = (float*)d_ws;
  float* G  = ws;                   // B*4096   = 16384 floats
  float* M2 = ws + 16384;           // B*512*64 = 131072 floats
  float* P  = ws + 16384 + 131072;  // B*4096   = 16384 floats

  k_zero<<<(16384 + 255) / 256, 256, 0, stream>>>(G, 16384);
  k_gram<<<B_DIM * 64, 512, 0, stream>>>(emb, G);
  k_attn<<<B_DIM * H_DIM, 512, 0, stream>>>(G, Wq, Wk, Wv, M2);
  k_pmat<<<B_DIM, 512, 0, stream>>>(M2, Wo, P);
  k_out<<<B_DIM * 64, 512, 0, stream>>>(emb, P, out);
}